// LongformerSelfAttention_23905787970009
// MI455X (gfx1250) — compile-verified
//
#include <hip/hip_runtime.h>
#include <hip/hip_bf16.h>
#include <math.h>

typedef float v2f __attribute__((ext_vector_type(2)));
typedef float v8f __attribute__((ext_vector_type(8)));

#define BATCH 32
#define SEQ   512
#define HID   1024
#define NHEAD 16
#define HDIM  64

// ---------------------------------------------------------------------------
// Kernel 1: C[M,N] = A[M,K] @ W[K,N] + bias, fp32 WMMA 16x16x4
// Block tile 64(M) x 128(N), BK=32, 256 thr = 8 waves, 32x32 (2x2 tiles)/wave
// LDS double-buffered: global loads for tile k+1 overlap WMMAs on tile k.
// ---------------------------------------------------------------------------
#define BM 64
#define BN 128
#define BK 32

__global__ __launch_bounds__(256) void gemm_bias_kernel(
    const float* __restrict__ A, const float* __restrict__ W,
    const float* __restrict__ bias, float* __restrict__ C,
    int M, int N, int K)
{
    __shared__ __align__(16) float As[2][BM][BK + 4];   // 2*64*36*4  = 18432 B
    __shared__ __align__(16) float Bs[2][BK][BN + 4];   // 2*32*132*4 = 33792 B

    const int tid  = threadIdx.x;
    const int lane = tid & 31;
    const int wave = tid >> 5;       // 0..7
    const int half = lane >> 4;      // 0/1 (lane group -> K pair select)
    const int l16  = lane & 15;

    const int rowTile = blockIdx.y * BM;
    const int colTile = blockIdx.x * BN;
    const int waveM   = (wave >> 2) * 32;   // 0 or 32
    const int waveN   = (wave & 3) * 32;    // 0,32,64,96

    v8f acc[2][2];
    #pragma unroll
    for (int i = 0; i < 2; ++i)
        #pragma unroll
        for (int j = 0; j < 2; ++j)
            acc[i][j] = (v8f){0.f, 0.f, 0.f, 0.f, 0.f, 0.f, 0.f, 0.f};

    // prologue: stage tile k0=0 into buffer 0
    #pragma unroll
    for (int j = 0; j < 2; ++j) {                   // A: 512 float4
        const int idx = tid + j * 256;
        const int ar  = idx >> 3, ac4 = (idx & 7) * 4;
        *(float4*)&As[0][ar][ac4] =
            *(const float4*)&A[(size_t)(rowTile + ar) * K + ac4];
    }
    #pragma unroll
    for (int j = 0; j < 4; ++j) {                   // B: 1024 float4
        const int idx = tid + j * 256;
        const int br  = idx >> 5, bc4 = (idx & 31) * 4;
        *(float4*)&Bs[0][br][bc4] =
            *(const float4*)&W[(size_t)br * N + colTile + bc4];
    }
    __syncthreads();

    for (int k0 = 0; k0 < K; k0 += BK) {
        const int buf  = (k0 >> 5) & 1;
        const int nbuf = buf ^ 1;
        const int kn   = k0 + BK;
        if (kn < K) {   // stage next tile into the other buffer
            #pragma unroll
            for (int j = 0; j < 2; ++j) {
                const int idx = tid + j * 256;
                const int ar  = idx >> 3, ac4 = (idx & 7) * 4;
                *(float4*)&As[nbuf][ar][ac4] =
                    *(const float4*)&A[(size_t)(rowTile + ar) * K + kn + ac4];
            }
            #pragma unroll
            for (int j = 0; j < 4; ++j) {
                const int idx = tid + j * 256;
                const int br  = idx >> 5, bc4 = (idx & 31) * 4;
                *(float4*)&Bs[nbuf][br][bc4] =
                    *(const float4*)&W[(size_t)(kn + br) * N + colTile + bc4];
            }
        }
        #pragma unroll
        for (int kk = 0; kk < BK; kk += 4) {
            const int ka = kk + 2 * half;   // lanes 0-15: K0/K1 ; 16-31: K2/K3
            v2f a0, a1, b0, b1;
            a0.x = As[buf][waveM + l16][ka];        a0.y = As[buf][waveM + l16][ka + 1];
            a1.x = As[buf][waveM + 16 + l16][ka];   a1.y = As[buf][waveM + 16 + l16][ka + 1];
            b0.x = Bs[buf][ka][waveN + l16];        b0.y = Bs[buf][ka + 1][waveN + l16];
            b1.x = Bs[buf][ka][waveN + 16 + l16];   b1.y = Bs[buf][ka + 1][waveN + 16 + l16];
            acc[0][0] = __builtin_amdgcn_wmma_f32_16x16x4_f32(false, a0, false, b0, (short)0, acc[0][0], false, false);
            acc[0][1] = __builtin_amdgcn_wmma_f32_16x16x4_f32(false, a0, false, b1, (short)0, acc[0][1], false, false);
            acc[1][0] = __builtin_amdgcn_wmma_f32_16x16x4_f32(false, a1, false, b0, (short)0, acc[1][0], false, false);
            acc[1][1] = __builtin_amdgcn_wmma_f32_16x16x4_f32(false, a1, false, b1, (short)0, acc[1][1], false, false);
        }
        __syncthreads();
    }

    #pragma unroll
    for (int i = 0; i < 2; ++i) {
        #pragma unroll
        for (int j = 0; j < 2; ++j) {
            const int col = colTile + waveN + j * 16 + l16;
            const float bcol = bias[col];
            #pragma unroll
            for (int r = 0; r < 8; ++r) {
                const int row = rowTile + waveM + i * 16 + r + 8 * half;
                C[(size_t)row * N + col] = acc[i][j][r] + bcol;
            }
        }
    }
}

// ---------------------------------------------------------------------------
// Kernel 2: local[b,t] = dot(Q[b,t,:64], K[b,t,:64]); one wave32 per row
// ---------------------------------------------------------------------------
__global__ __launch_bounds__(256) void local_dot_kernel(
    const float* __restrict__ Q, const float* __restrict__ K,
    float* __restrict__ localOut, int nrows)
{
    const int wave = threadIdx.x >> 5;
    const int lane = threadIdx.x & 31;
    const int row  = blockIdx.x * 8 + wave;
    if (row >= nrows) return;
    const float2 q2 = *(const float2*)&Q[(size_t)row * HDIM + lane * 2];
    const float2 k2 = *(const float2*)&K[(size_t)row * HDIM + lane * 2];
    float v = q2.x * k2.x + q2.y * k2.y;
    #pragma unroll
    for (int off = 16; off > 0; off >>= 1)
        v += __shfl_xor(v, off, 32);
    if (lane == 0) localOut[row] = v;
}

// ---------------------------------------------------------------------------
// Kernel 3: per (b,n): G = Wp^T(512x1536) @ X(1536x64), X = [Qn;Kn;Vn]
// then out = p0*V + (1-p0)*gelu(G + bp), p0 = sigmoid(local/8)
// Block: 256(s') x 64(d) tile, K=1536 in 32-steps; each wave: 2x4 WMMA tiles
// ---------------------------------------------------------------------------
#define GM 256
#define GN 64
#define GK 32

__global__ __launch_bounds__(256) void gv_ctx_kernel(
    const float* __restrict__ Q, const float* __restrict__ Kx,
    const float* __restrict__ V, const float* __restrict__ Wp,
    const float* __restrict__ bp, const float* __restrict__ localv,
    float* __restrict__ out)
{
    __shared__ __align__(16) float As[GK][GM + 4];   // 32*260*4 = 33280 B
    __shared__ __align__(16) float Bs[GK][GN + 4];   // 32*68*4  =  8704 B

    const int tid  = threadIdx.x;
    const int lane = tid & 31;
    const int wave = tid >> 5;     // wave covers rows [wave*32, wave*32+32)
    const int half = lane >> 4;
    const int l16  = lane & 15;

    const int bn = blockIdx.y;     // 0..511
    const int b  = bn >> 4;
    const int n  = bn & 15;
    const int sBase = blockIdx.x * GM;

    const size_t qkvOff = (size_t)b * (SEQ * HID) + (size_t)n * (SEQ * HDIM);

    v8f acc[2][4];
    #pragma unroll
    for (int i = 0; i < 2; ++i)
        #pragma unroll
        for (int c = 0; c < 4; ++c)
            acc[i][c] = (v8f){0.f, 0.f, 0.f, 0.f, 0.f, 0.f, 0.f, 0.f};

    for (int k0 = 0; k0 < 3 * SEQ; k0 += GK) {
        const float* X = (k0 < SEQ) ? Q : (k0 < 2 * SEQ) ? Kx : V;
        const int kloc = k0 & (SEQ - 1);

        #pragma unroll
        for (int j = 0; j < 8; ++j) {           // Wp^T tile: 2048 float4
            const int idx = tid + j * 256;
            const int kr  = idx >> 6;           // 0..31
            const int c4  = (idx & 63) * 4;     // 0..252
            *(float4*)&As[kr][c4] =
                *(const float4*)&Wp[(size_t)(k0 + kr) * SEQ + sBase + c4];
        }
        #pragma unroll
        for (int j = 0; j < 2; ++j) {           // X tile: 512 float4
            const int idx = tid + j * 256;
            const int kr  = idx >> 4;           // 0..31
            const int c4  = (idx & 15) * 4;     // 0..60
            *(float4*)&Bs[kr][c4] =
                *(const float4*)&X[qkvOff + (size_t)(kloc + kr) * HDIM + c4];
        }
        __syncthreads();

        #pragma unroll
        for (int kk = 0; kk < GK; kk += 4) {
            const int ka = kk + 2 * half;
            v2f a0, a1;
            a0.x = As[ka][wave * 32 + l16];       a0.y = As[ka + 1][wave * 32 + l16];
            a1.x = As[ka][wave * 32 + 16 + l16];  a1.y = As[ka + 1][wave * 32 + 16 + l16];
            #pragma unroll
            for (int c = 0; c < 4; ++c) {
                v2f bfr;
                bfr.x = Bs[ka][c * 16 + l16];
                bfr.y = Bs[ka + 1][c * 16 + l16];
                acc[0][c] = __builtin_amdgcn_wmma_f32_16x16x4_f32(false, a0, false, bfr, (short)0, acc[0][c], false, false);
                acc[1][c] = __builtin_amdgcn_wmma_f32_16x16x4_f32(false, a1, false, bfr, (short)0, acc[1][c], false, false);
            }
        }
        __syncthreads();
    }

    // fused epilogue: bias + gelu + 2-way softmax blend + transposed store
    const float kInvSqrt2 = 1.0f / 1.41421f;   // matches reference erf(x/1.41421)
    #pragma unroll
    for (int rt = 0; rt < 2; ++rt) {
        #pragma unroll
        for (int r = 0; r < 8; ++r) {
            const int srow = sBase + wave * 32 + rt * 16 + r + 8 * half;
            const float x  = localv[(size_t)b * (NHEAD * SEQ) + n * SEQ + srow] * 0.125f;
            const float p0 = 1.0f / (1.0f + expf(-x));   // softmax([x,0])[0]
            const float p1 = 1.0f - p0;
            const float bpv = bp[srow];
            const size_t vbase = qkvOff + (size_t)srow * HDIM;
            const size_t obase = (size_t)b * (SEQ * HID) + (size_t)srow * HID + n * HDIM;
            #pragma unroll
            for (int c = 0; c < 4; ++c) {
                const int d = c * 16 + l16;
                const float g  = acc[rt][c][r] + bpv;
                const float ge = 0.5f * g * (1.0f + erff(g * kInvSqrt2));
                out[obase + d] = p0 * V[vbase + d] + p1 * ge;
            }
        }
    }
}

// ---------------------------------------------------------------------------
// Launch
// ---------------------------------------------------------------------------
extern "C" void kernel_launch(void* const* d_in, const int* in_sizes, int n_in,
                              void* d_out, int out_size, void* d_ws, size_t ws_size,
                              hipStream_t stream) {
    const float* hs = (const float*)d_in[0];
    const float* Wq = (const float*)d_in[1];
    const float* bq = (const float*)d_in[2];
    const float* Wk = (const float*)d_in[3];
    const float* bk = (const float*)d_in[4];
    const float* Wv = (const float*)d_in[5];
    const float* bv = (const float*)d_in[6];
    const float* Wp = (const float*)d_in[7];
    const float* bp = (const float*)d_in[8];
    float* out = (float*)d_out;

    const int M = BATCH * SEQ;                 // 16384
    const size_t qkvElems = (size_t)M * HID;   // 16.7M floats each

    float* Q  = (float*)d_ws;
    float* K  = Q + qkvElems;
    float* V  = K + qkvElems;
    float* lv = V + qkvElems;                  // local scores: B*NH*S floats

    dim3 blk(256);
    dim3 ggrid(HID / BN, M / BM);              // (8, 256)
    gemm_bias_kernel<<<ggrid, blk, 0, stream>>>(hs, Wq, bq, Q, M, HID, HID);
    gemm_bias_kernel<<<ggrid, blk, 0, stream>>>(hs, Wk, bk, K, M, HID, HID);
    gemm_bias_kernel<<<ggrid, blk, 0, stream>>>(hs, Wv, bv, V, M, HID, HID);

    const int nrows = BATCH * NHEAD * SEQ;     // 262144
    local_dot_kernel<<<nrows / 8, blk, 0, stream>>>(Q, K, lv, nrows);

    dim3 ggrid2(SEQ / GM, BATCH * NHEAD);      // (2, 512)
    gv_ctx_kernel<<<ggrid2, blk, 0, stream>>>(Q, K, V, Wp, bp, lv, out);
}